// PyGraph_56143812493354
// MI455X (gfx1250) — compile-verified
//
#include <hip/hip_runtime.h>
#include <hip/hip_bf16.h>

typedef float v2f __attribute__((ext_vector_type(2)));
typedef float v8f __attribute__((ext_vector_type(8)));
typedef unsigned int u32x4 __attribute__((ext_vector_type(4)));
typedef int i32x4 __attribute__((ext_vector_type(4)));
typedef int i32x8 __attribute__((ext_vector_type(8)));

#define NN       8192
#define CDIM     64
#define KNN      9
#define QB       128      // queries per knn block
#define CH       64       // candidate columns per chunk
#define NCH      (1024 / CH)
#define LSTRIDE  68       // LDS row stride in floats (16B aligned, conflict-spread)
#define CBUFB    (CH * LSTRIDE * 4)      // bytes per candidate buffer

#if __has_builtin(__builtin_amdgcn_tensor_load_to_lds)
#define HAVE_TDM 1
#else
#define HAVE_TDM 0
#endif

__device__ __forceinline__ int batch_of(int i) {
    // matches jnp.linspace(0, 8, 8192).astype(int32)
    return (int)((float)i * (8.0f / 8191.0f));
}

__device__ __forceinline__ void wait_tensorcnt_0() {
#if __has_builtin(__builtin_amdgcn_s_wait_tensorcnt)
    __builtin_amdgcn_s_wait_tensorcnt(0);
#else
    asm volatile("s_wait_tensorcnt 0x0" ::: "memory");
#endif
}
__device__ __forceinline__ void wait_tensorcnt_1() {
#if __has_builtin(__builtin_amdgcn_s_wait_tensorcnt)
    __builtin_amdgcn_s_wait_tensorcnt(1);
#else
    asm volatile("s_wait_tensorcnt 0x1" ::: "memory");
#endif
}

#if HAVE_TDM
// Issue a TDM load of a 64-row x 64-col f32 tile (row stride 64 elements in
// memory) into LDS, with hardware padding of 4 DWORDs after every 64 DWORDs
// so the LDS image has the 68-float row stride the WMMA fragments expect.
__device__ __forceinline__ void tdm_load_tile(void* lds_dst, const void* gsrc) {
    unsigned lds = (unsigned)(uintptr_t)lds_dst;            // LDS byte offset
    unsigned long long ga = (unsigned long long)(uintptr_t)gsrc;

    u32x4 g0;
    g0[0] = 1u;                                             // count=1 (valid user D#)
    g0[1] = lds;                                            // lds_addr
    g0[2] = (unsigned)(ga & 0xFFFFFFFFu);                   // global_addr[31:0]
    g0[3] = (unsigned)((ga >> 32) & 0x01FFFFFFu)            // global_addr[56:32]
          | (2u << 30);                                     // type=2 ("image")

    i32x8 g1;
    g1[0] = (int)((2u << 16)      // data_size = 4B
                | (1u << 20)      // pad_enable
                | (5u << 22)      // pad_interval: 64 DWORDs
                | (3u << 25));    // pad_amount: 4 DWORDs
    g1[1] = (int)(64u << 16);     // tensor_dim0[15:0] = 64 (row length, elems)
    g1[2] = (int)(1024u << 16);   // tensor_dim1[15:0] = 1024 (rows available)
    g1[3] = (int)(64u << 16);     // tile_dim0 = 64
    g1[4] = (int)64;              // tile_dim1 = 64, tile_dim2 = 0
    g1[5] = (int)64;              // tensor_dim0_stride[31:0] = 64 elems
    g1[6] = 0;
    g1[7] = 0;

    i32x4 z4 = {0, 0, 0, 0};
#if defined(__clang_major__) && __clang_major__ >= 23
    i32x8 z8 = {0, 0, 0, 0, 0, 0, 0, 0};
    __builtin_amdgcn_tensor_load_to_lds(g0, g1, z4, z4, z8, 0);
#else
    __builtin_amdgcn_tensor_load_to_lds(g0, g1, z4, z4, 0);
#endif
}
#endif

// ---------------------------------------------------------------------------
// Kernel 1: pack x[B,C,H,W] -> xf[N,C] (row-major nodes) + sq[n] = |x_n|^2
// ---------------------------------------------------------------------------
__global__ void __launch_bounds__(256)
k_pack(const float* __restrict__ x, float* __restrict__ xf, float* __restrict__ sq) {
    __shared__ float tile[64][65];
    __shared__ float psum[64][4];
    int blk = blockIdx.x;            // 128 blocks: 8 batches x 16 hw-tiles
    int b   = blk >> 4;
    int hw0 = (blk & 15) << 6;
    int t   = threadIdx.x;
    int cl  = t >> 6;                // 0..3
    int hwl = t & 63;

    #pragma unroll
    for (int i = 0; i < 16; ++i) {
        int c = (i << 2) + cl;
        tile[c][hwl] = x[((b * 64 + c) << 10) + hw0 + hwl];
    }
    __syncthreads();

    int c  = t & 63;
    int nl = t >> 6;
    #pragma unroll
    for (int i = 0; i < 16; ++i) {
        int r = (i << 2) + nl;
        xf[(((b << 10) + hw0 + r) << 6) + c] = tile[c][r];
    }

    float s = 0.f;
    #pragma unroll
    for (int i = 0; i < 16; ++i) {
        float v = tile[cl * 16 + i][hwl];
        s += v * v;
    }
    psum[hwl][cl] = s;
    __syncthreads();
    if (cl == 0)
        sq[(b << 10) + hw0 + hwl] = psum[hwl][0] + psum[hwl][1] + psum[hwl][2] + psum[hwl][3];
}

// ---------------------------------------------------------------------------
// Kernel 2: KNN (top-9 by squared distance within batch segment, self incl.)
// score(q,c) = |c|^2 - 2 q.c   (row-constant |q|^2 dropped: same ordering)
// Candidate tiles streamed by the Tensor Data Mover, double buffered on
// TENSORcnt so DMA of chunk ch+1 overlaps WMMA+selection of chunk ch.
// ---------------------------------------------------------------------------
__global__ void __launch_bounds__(256)
k_knn(const float* __restrict__ xf, const float* __restrict__ sq,
      int* __restrict__ knn_idx) {
    extern __shared__ char smem[];
    float* buf0     = (float*)smem;
    float* buf1     = (float*)(smem + CBUFB);
    float* sScore   = (float*)(smem + 2 * CBUFB);                    // QB x LSTRIDE
    float* sMrgD    = (float*)(smem + 2 * CBUFB + QB * LSTRIDE * 4); // QB x 9
    int*   sMrgI    = (int*)  (smem + 2 * CBUFB + QB * LSTRIDE * 4 + QB * KNN * 4);

    int t    = threadIdx.x;
    int q0   = blockIdx.x * QB;
    int s0   = q0 & ~1023;                    // batch segment start (1024 wide)
    int wave = t >> 5;
    int lane = t & 31;
    int m16  = lane & 15;
    int kh   = (lane >> 4) << 1;              // 0 or 2 (K sub-offset per half-wave)
    int rowbase = q0 + wave * 16;             // this wave's 16 query rows

    // A fragments for this wave's 16 queries, all K=64 (16 steps of 4)
    v2f a[16];
    {
        const float* row = xf + (rowbase + m16) * CDIM;
        #pragma unroll
        for (int kk = 0; kk < 16; ++kk) {
            a[kk].x = row[kk * 4 + kh];
            a[kk].y = row[kk * 4 + kh + 1];
        }
    }

    // per-thread running top-9 (2 threads per query row)
    int   myrow = t >> 1;
    int   half  = t & 1;
    int   q     = q0 + myrow;
    int   bq    = batch_of(q);
    float bd[KNN];
    int   bi[KNN];
    #pragma unroll
    for (int k = 0; k < KNN; ++k) { bd[k] = __builtin_inff(); bi[k] = q; }

#if HAVE_TDM
    if (t == 0) tdm_load_tile(buf0, xf + (size_t)s0 * CDIM);   // prefetch chunk 0
#endif

    for (int ch = 0; ch < NCH; ++ch) {
        int c0 = s0 + ch * CH;
        float* cur = (ch & 1) ? buf1 : buf0;
        float* nxt = (ch & 1) ? buf0 : buf1;
        __syncthreads();   // prior compute done: safe to overwrite 'nxt'
#if HAVE_TDM
        if (t == 0) {
            if (ch + 1 < NCH) {
                tdm_load_tile(nxt, xf + (size_t)(c0 + CH) * CDIM);
                wait_tensorcnt_1();   // oldest (chunk ch) complete, ch+1 in flight
            } else {
                wait_tensorcnt_0();
            }
        }
#else
        {   // fallback: cooperative staging
            int rl  = t >> 2;
            int cb4 = (t & 3) * 16;
            const float* src = xf + (c0 + rl) * CDIM + cb4;
            float*       dst = cur + rl * LSTRIDE + cb4;
            #pragma unroll
            for (int i = 0; i < 16; ++i) dst[i] = src[i];
        }
#endif
        __syncthreads();
        const float* sXc = cur;

        // WMMA: each wave -> 4 column tiles of 16 candidates
        #pragma unroll
        for (int j = 0; j < CH / 16; ++j) {
            v8f acc = {};
            #pragma unroll
            for (int kk = 0; kk < 16; ++kk) {
                v2f bf;
                const float* bp = sXc + (j * 16 + m16) * LSTRIDE + kk * 4 + kh;
                bf.x = bp[0];
                bf.y = bp[1];
                acc = __builtin_amdgcn_wmma_f32_16x16x4_f32(
                        false, a[kk], false, bf, (short)0, acc, false, false);
            }
            // fold |c|^2 and write scores: row = wave*16 + v + 8*(lane/16)
            float sqc  = sq[c0 + j * 16 + m16];
            int   rloc = wave * 16 + ((lane >> 4) << 3);
            float* outp = sScore + rloc * LSTRIDE + j * 16 + m16;
            #pragma unroll
            for (int v = 0; v < 8; ++v) outp[v * LSTRIDE] = sqc - 2.0f * acc[v];
        }
        __syncthreads();

        // selection: thread (myrow, half) scans 32 columns
        {
            const float* srow  = sScore + myrow * LSTRIDE + half * 32;
            int          cbase = c0 + half * 32;
            for (int j = 0; j < 32; ++j) {
                int cand = cbase + j;
                if (batch_of(cand) != bq) continue;
                float sc = srow[j];
                if (sc < bd[KNN - 1]) {
                    int p = KNN - 1;
                    while (p > 0 && bd[p - 1] > sc) {
                        bd[p] = bd[p - 1]; bi[p] = bi[p - 1]; --p;
                    }
                    bd[p] = sc; bi[p] = cand;
                }
            }
        }
        // merge odd-half list into even-half running list
        if (half == 1) {
            #pragma unroll
            for (int k = 0; k < KNN; ++k) {
                sMrgD[myrow * KNN + k] = bd[k];
                sMrgI[myrow * KNN + k] = bi[k];
                bd[k] = __builtin_inff(); bi[k] = q;
            }
        }
        __syncthreads();
        if (half == 0) {
            #pragma unroll
            for (int k = 0; k < KNN; ++k) {
                float sc = sMrgD[myrow * KNN + k];
                if (sc < bd[KNN - 1]) {
                    int id = sMrgI[myrow * KNN + k];
                    int p = KNN - 1;
                    while (p > 0 && bd[p - 1] > sc) {
                        bd[p] = bd[p - 1]; bi[p] = bi[p - 1]; --p;
                    }
                    bd[p] = sc; bi[p] = id;
                }
            }
        }
    }
    __syncthreads();
    if (half == 0) {
        #pragma unroll
        for (int k = 0; k < KNN; ++k) knn_idx[q * KNN + k] = bi[k];
    }
}

// ---------------------------------------------------------------------------
// Kernel 3a/3b: degree over src (self loops excluded)
// ---------------------------------------------------------------------------
__global__ void k_zero(int* __restrict__ p, int n) {
    int i = blockIdx.x * blockDim.x + threadIdx.x;
    if (i < n) p[i] = 0;
}

__global__ void k_deg(const int* __restrict__ knn_idx, int* __restrict__ deg) {
    int e = blockIdx.x * blockDim.x + threadIdx.x;
    if (e >= NN * KNN) return;
    int s = knn_idx[e];
    int t = e / KNN;
    if (s != t) atomicAdd(&deg[s], 1);
}

// ---------------------------------------------------------------------------
// Kernel 4: fused gather (Tx1) + out = X@W0 + Tx1@W1 + bias   (WMMA f32)
// ---------------------------------------------------------------------------
__global__ void __launch_bounds__(128)
k_out(const float* __restrict__ xf, const int* __restrict__ knn_idx,
      const int* __restrict__ deg, const float* __restrict__ W0,
      const float* __restrict__ W1, const float* __restrict__ bias,
      float* __restrict__ out) {
    __shared__ float sX[64 * LSTRIDE];
    __shared__ float sT[64 * LSTRIDE];
    __shared__ float sW[64 * 64];

    int t  = threadIdx.x;
    int n0 = blockIdx.x * 64;
    int nl = t >> 1;
    int hf = t & 1;
    int n  = n0 + nl;
    int cb = hf * 32;

    // gather stage: Tx1 row for node n (this thread: 32 channels)
    int   dg = deg[n];
    float dn = dg > 0 ? rsqrtf((float)dg) : 0.0f;
    float acc[32];
    #pragma unroll
    for (int i = 0; i < 32; ++i) acc[i] = 0.f;
    for (int k = 0; k < KNN; ++k) {
        int   s  = knn_idx[n * KNN + k];
        int   ds = deg[s];
        float w  = (s != n && ds > 0) ? (-dn * rsqrtf((float)ds)) : 0.0f;
        const float* xs = xf + s * CDIM + cb;
        #pragma unroll
        for (int i = 0; i < 32; i += 4) {
            float4 v = *(const float4*)(xs + i);
            acc[i]     += w * v.x;
            acc[i + 1] += w * v.y;
            acc[i + 2] += w * v.z;
            acc[i + 3] += w * v.w;
        }
    }
    {
        const float* xn   = xf + n * CDIM + cb;
        float*       dstT = sT + nl * LSTRIDE + cb;
        float*       dstX = sX + nl * LSTRIDE + cb;
        #pragma unroll
        for (int i = 0; i < 32; ++i) { dstT[i] = acc[i]; dstX[i] = xn[i]; }
    }
    for (int i = t; i < 4096; i += 128) sW[i] = W0[i];
    __syncthreads();

    int wave = t >> 5, lane = t & 31;
    int m16  = lane & 15, kh = (lane >> 4) << 1;

    // A fragments for this wave's 16-row tile (X and Tx1)
    v2f ax[16], at_[16];
    {
        const float* rx = sX + (wave * 16 + m16) * LSTRIDE;
        const float* rt = sT + (wave * 16 + m16) * LSTRIDE;
        #pragma unroll
        for (int kk = 0; kk < 16; ++kk) {
            int c = kk * 4 + kh;
            ax[kk].x  = rx[c]; ax[kk].y  = rx[c + 1];
            at_[kk].x = rt[c]; at_[kk].y = rt[c + 1];
        }
    }
    v8f accd[4];
    #pragma unroll
    for (int j = 0; j < 4; ++j) {
        float bv = bias[j * 16 + m16];     // D column = j*16 + lane%16
        #pragma unroll
        for (int v = 0; v < 8; ++v) accd[j][v] = bv;
    }
    // pass 0: + X @ W0
    #pragma unroll
    for (int j = 0; j < 4; ++j) {
        #pragma unroll
        for (int kk = 0; kk < 16; ++kk) {
            v2f b;
            const float* bp = sW + (kk * 4 + kh) * 64 + j * 16 + m16;
            b.x = bp[0];
            b.y = bp[64];
            accd[j] = __builtin_amdgcn_wmma_f32_16x16x4_f32(
                        false, ax[kk], false, b, (short)0, accd[j], false, false);
        }
    }
    __syncthreads();
    for (int i = t; i < 4096; i += 128) sW[i] = W1[i];
    __syncthreads();
    // pass 1: + Tx1 @ W1
    #pragma unroll
    for (int j = 0; j < 4; ++j) {
        #pragma unroll
        for (int kk = 0; kk < 16; ++kk) {
            v2f b;
            const float* bp = sW + (kk * 4 + kh) * 64 + j * 16 + m16;
            b.x = bp[0];
            b.y = bp[64];
            accd[j] = __builtin_amdgcn_wmma_f32_16x16x4_f32(
                        false, at_[kk], false, b, (short)0, accd[j], false, false);
        }
    }
    // store: row = n0 + wave*16 + v + 8*(lane/16), col = j*16 + lane%16
    {
        int rbase = n0 + wave * 16 + ((lane >> 4) << 3);
        #pragma unroll
        for (int j = 0; j < 4; ++j) {
            float* op = out + rbase * 64 + j * 16 + m16;
            #pragma unroll
            for (int v = 0; v < 8; ++v) op[v * 64] = accd[j][v];
        }
    }
}

// ---------------------------------------------------------------------------
extern "C" void kernel_launch(void* const* d_in, const int* in_sizes, int n_in,
                              void* d_out, int out_size, void* d_ws, size_t ws_size,
                              hipStream_t stream) {
    const float* x    = (const float*)d_in[0];
    const float* W0   = (const float*)d_in[1];
    const float* W1   = (const float*)d_in[2];
    const float* bias = (const float*)d_in[3];
    float*       out  = (float*)d_out;

    char*  ws      = (char*)d_ws;
    float* xf      = (float*)(ws);                                   // 2 MB
    float* sq      = (float*)(ws + 2u * 1024 * 1024);                // 32 KB
    int*   knn_idx = (int*)  (ws + 2u * 1024 * 1024 + 32 * 1024);    // 288 KB
    int*   deg     = (int*)  (ws + 2u * 1024 * 1024 + 32 * 1024 + NN * KNN * 4);

    size_t knn_lds = 2 * CBUFB + QB * LSTRIDE * 4 + QB * KNN * 8;    // ~77 KB

    k_pack<<<128, 256, 0, stream>>>(x, xf, sq);
    k_zero<<<(NN + 255) / 256, 256, 0, stream>>>(deg, NN);
    k_knn <<<NN / QB, 256, knn_lds, stream>>>(xf, sq, knn_idx);
    k_deg <<<(NN * KNN + 255) / 256, 256, 0, stream>>>(knn_idx, deg);
    k_out <<<NN / 64, 128, 0, stream>>>(xf, knn_idx, deg, W0, W1, bias, out);
}